// FASTAModel2_67276367724916
// MI455X (gfx1250) — compile-verified
//
#include <hip/hip_runtime.h>
#include <hip/hip_bf16.h>
#include <math.h>

typedef float v2f __attribute__((ext_vector_type(2)));
typedef float v8f __attribute__((ext_vector_type(8)));

#define D_MODEL 166
#define N_LAYER 3
#define D_INNER 332
#define D_STATE 16
#define D_CONV  4
#define DT_RANK 11
#define BATCH   16
#define SEQ     1024
#define BL      (BATCH*SEQ)
#define DBC_W   (DT_RANK + 2*D_STATE)   // 43
#define LCH     64                       // scan chunk length
#define NCHUNK  (SEQ/LCH)                // 16 chunks

// padded/transposed weight sizes
#define IN_NP  672
#define IN_KP  168
#define XP_NP  48
#define XP_KP  332
#define DT_NP  336
#define DT_KP  12
#define OP_NP  176
#define OP_KP  332

#if __has_builtin(__builtin_amdgcn_s_wait_asynccnt)
#define WAIT_ASYNC0() __builtin_amdgcn_s_wait_asynccnt(0)
#else
#define WAIT_ASYNC0() asm volatile("s_wait_asynccnt 0x0" ::: "memory")
#endif

// ---------------- embedding gather ----------------
__global__ void embed_kernel(const int* __restrict__ fasta, const float* __restrict__ emb,
                             float* __restrict__ hidden) {
  int idx = blockIdx.x * blockDim.x + threadIdx.x;
  if (idx >= BL * D_MODEL) return;
  int row = idx / D_MODEL, c = idx % D_MODEL;
  hidden[idx] = emb[fasta[row] * D_MODEL + c];
}

// ---- weight pad + transpose, k-pair interleaved: Wt2[((k/2)*Npad+n)*2+(k&1)] = W[n*K+k]
// Each lane's (k, k+1) B pair is one aligned 8B chunk; consecutive lanes are consecutive
// chunks -> single coalesced global_load_b64 per k-step in the GEMM.
__global__ void padT2_kernel(const float* __restrict__ W, float* __restrict__ Wt2,
                             int N, int K, int Npad, int Kpad) {
  int idx = blockIdx.x * blockDim.x + threadIdx.x;
  if (idx >= Npad * Kpad) return;
  int e = idx & 1, pn = idx >> 1;
  int p = pn / Npad, n = pn % Npad;
  int k = p * 2 + e;
  Wt2[idx] = (n < N && k < K) ? W[(size_t)n * K + k] : 0.f;
}

// ---------------- residual accumulate + RMSNorm (one wave per row) ----------------
__global__ void resid_rmsnorm_kernel(const float* __restrict__ h, float* __restrict__ resid,
                                     const float* __restrict__ w, float* __restrict__ outN,
                                     int first) {
  int wave = threadIdx.x >> 5, lane = threadIdx.x & 31;
  int row = blockIdx.x * (blockDim.x >> 5) + wave;
  if (row >= BL) return;
  const float* hr = h + (size_t)row * D_MODEL;
  float* rr = resid + (size_t)row * D_MODEL;
  float ss = 0.f;
  for (int i = lane; i < D_MODEL; i += 32) {
    float v = first ? hr[i] : (rr[i] + hr[i]);
    rr[i] = v;
    ss += v * v;
  }
#pragma unroll
  for (int o = 16; o; o >>= 1) ss += __shfl_xor(ss, o, 32);
  float inv = rsqrtf(ss * (1.f / D_MODEL) + 1e-5f);
  for (int i = lane; i < D_MODEL; i += 32)
    outN[(size_t)row * D_MODEL + i] = rr[i] * inv * w[i];
}

// ---------------- fp32 WMMA GEMM: C[M,N] = A[M,K] * W^T via padded Wt2 ----------------
// Guard-free, coalesced b64 loads for A and B; software-pipelined; one 16x16 tile/wave.
// epi==1: C = softplus(acc + bias[n])   (dt_proj fusion)
__global__ void wmma_gemm_kernel(const float* __restrict__ Amat, int lda,
                                 const float* __restrict__ Wt2, int Npad,
                                 float* __restrict__ Cmat,
                                 int N, int Kpad, int Nt, int totalTiles,
                                 const float* __restrict__ bias, int epi) {
  int wave = threadIdx.x >> 5, lane = threadIdx.x & 31;
  int tile = blockIdx.x * (blockDim.x >> 5) + wave;
  if (tile >= totalTiles) return;           // wave-uniform guard: EXEC all-ones at WMMA
  int mt = tile / Nt, nt = tile % Nt;
  int m0 = mt << 4, n0 = nt << 4;
  int half = lane >> 4, r15 = lane & 15;
  int nc = n0 + r15;

  const float* Ap = Amat + (size_t)(m0 + r15) * lda + 2 * half;
  const float* Bp = Wt2 + ((size_t)half * Npad + nc) * 2;   // 8B-aligned pair
  size_t bstep = (size_t)4 * Npad;                          // advance 4 k per step

  v8f acc = {0.f, 0.f, 0.f, 0.f, 0.f, 0.f, 0.f, 0.f};
  v2f a, b;
  a.x = Ap[0]; a.y = Ap[1];
  b = *(const v2f*)Bp;
  for (int k = 4; k < Kpad; k += 4) {
    Ap += 4; Bp += bstep;
    v2f a2, b2;
    a2.x = Ap[0]; a2.y = Ap[1];
    b2 = *(const v2f*)Bp;
    __builtin_prefetch(Ap + 32, 0, 3);
    acc = __builtin_amdgcn_wmma_f32_16x16x4_f32(false, a, false, b,
                                                (short)0, acc, false, false);
    a = a2; b = b2;
  }
  acc = __builtin_amdgcn_wmma_f32_16x16x4_f32(false, a, false, b,
                                              (short)0, acc, false, false);

  if (nc < N) {
    float bv = (epi == 1) ? bias[nc] : 0.f;
#pragma unroll
    for (int r = 0; r < 8; r++) {
      int mrow = m0 + r + 8 * half;            // documented C/D layout
      float v = acc[r];
      if (epi == 1) { v += bv; v = (v > 20.f) ? v : log1pf(expf(v)); }
      Cmat[(size_t)mrow * N + nc] = v;
    }
  }
}

// ---------------- causal depthwise conv (width 4) + SiLU ----------------
__global__ void conv_silu_kernel(const float* __restrict__ xz, float* __restrict__ xc,
                                 const float* __restrict__ cw, const float* __restrict__ cb) {
  int idx = blockIdx.x * blockDim.x + threadIdx.x;
  if (idx >= BL * D_INNER) return;
  int d = idx % D_INNER;
  int l = (idx / D_INNER) % SEQ;
  int b = idx / (D_INNER * SEQ);
  float acc = cb[d];
#pragma unroll
  for (int j = 0; j < D_CONV; j++) {
    int ll = l - (D_CONV - 1) + j;
    if (ll >= 0)
      acc += xz[(size_t)(b * SEQ + ll) * (2 * D_INNER) + d] * cw[d * D_CONV + j];
  }
  float s = 1.f / (1.f + expf(-acc));
  xc[idx] = acc * s;   // silu
}

// ======== chunked parallel selective scan (h_t = a_t h_{t-1} + u_t is linear) ========
// Phase 1: per chunk, decay product P[n] and zero-init partial hp[n].  B*NCHUNK blocks.
__global__ void scan_phase1_kernel(const float* __restrict__ dtv, const float* __restrict__ xc,
                                   const float* __restrict__ dbc, const float* __restrict__ A_log,
                                   float* __restrict__ Pbuf, float* __restrict__ Hbuf) {
  __shared__ float sB[D_STATE];
  int b = blockIdx.x / NCHUNK, c = blockIdx.x % NCHUNK;
  int d = threadIdx.x;
  bool active = d < D_INNER;

  float Areg[D_STATE], P[D_STATE], hp[D_STATE];
  if (active) {
#pragma unroll
    for (int n = 0; n < D_STATE; n++) {
      Areg[n] = -expf(A_log[d * D_STATE + n]);
      P[n] = 1.f; hp[n] = 0.f;
    }
  }
  for (int tt = 0; tt < LCH; tt++) {
    size_t rowt = (size_t)b * SEQ + c * LCH + tt;
    if (threadIdx.x < D_STATE) {       // async-copy B vector -> LDS
      const float* gp = dbc + rowt * DBC_W + DT_RANK + threadIdx.x;
      unsigned lo = (unsigned)(size_t)&sB[threadIdx.x];
      asm volatile("global_load_async_to_lds_b32 %0, %1, off"
                   :: "v"(lo), "v"(gp) : "memory");
      WAIT_ASYNC0();
    }
    __syncthreads();
    if (active) {
      float dt = dtv[rowt * D_INNER + d];
      float dtx = dt * xc[rowt * D_INNER + d];
#pragma unroll
      for (int n = 0; n < D_STATE; n++) {
        float dA = expf(dt * Areg[n]);
        P[n] *= dA;
        hp[n] = dA * hp[n] + dtx * sB[n];
      }
    }
    __syncthreads();
  }
  if (active) {
    size_t off = ((size_t)(b * NCHUNK + c) * D_INNER + d) * D_STATE;
#pragma unroll
    for (int n = 0; n < D_STATE; n++) { Pbuf[off + n] = P[n]; Hbuf[off + n] = hp[n]; }
  }
}

// Phase 2: sequential combine over the 16 chunks; Hbuf becomes per-chunk initial state.
__global__ void scan_phase2_kernel(const float* __restrict__ Pbuf, float* __restrict__ Hbuf) {
  int idx = blockIdx.x * blockDim.x + threadIdx.x;
  if (idx >= BATCH * D_INNER * D_STATE) return;
  int b = idx / (D_INNER * D_STATE);
  int rem = idx % (D_INNER * D_STATE);
  float h = 0.f;
  for (int c = 0; c < NCHUNK; c++) {
    size_t off = (size_t)(b * NCHUNK + c) * D_INNER * D_STATE + rem;
    float hp = Hbuf[off];
    float P  = Pbuf[off];
    Hbuf[off] = h;             // initial state entering chunk c
    h = hp + P * h;            // exact linear composition
  }
}

// Phase 3: replay each chunk from corrected init; emit y = (h·C + D*x) * silu(z) in place.
__global__ void scan_phase3_kernel(const float* __restrict__ dtv, float* __restrict__ xy,
                                   const float* __restrict__ xz, const float* __restrict__ dbc,
                                   const float* __restrict__ A_log, const float* __restrict__ Dp,
                                   const float* __restrict__ Hbuf) {
  __shared__ float sBC[2 * D_STATE];   // [0:16)=B, [16:32)=C
  int b = blockIdx.x / NCHUNK, c = blockIdx.x % NCHUNK;
  int d = threadIdx.x;
  bool active = d < D_INNER;

  float Areg[D_STATE], h[D_STATE];
  float Dd = 0.f;
  if (active) {
    Dd = Dp[d];
    size_t off = ((size_t)(b * NCHUNK + c) * D_INNER + d) * D_STATE;
#pragma unroll
    for (int n = 0; n < D_STATE; n++) {
      Areg[n] = -expf(A_log[d * D_STATE + n]);
      h[n] = Hbuf[off + n];
    }
  }
  for (int tt = 0; tt < LCH; tt++) {
    size_t rowt = (size_t)b * SEQ + c * LCH + tt;
    if (threadIdx.x < 2 * D_STATE) {   // async-copy B and C vectors -> LDS
      const float* gp = dbc + rowt * DBC_W + DT_RANK + threadIdx.x;
      unsigned lo = (unsigned)(size_t)&sBC[threadIdx.x];
      asm volatile("global_load_async_to_lds_b32 %0, %1, off"
                   :: "v"(lo), "v"(gp) : "memory");
      WAIT_ASYNC0();
    }
    __syncthreads();
    if (active) {
      float dt = dtv[rowt * D_INNER + d];
      float xv = xy[rowt * D_INNER + d];
      float dtx = dt * xv;
      float y = 0.f;
#pragma unroll
      for (int n = 0; n < D_STATE; n++) {
        float dA = expf(dt * Areg[n]);
        h[n] = dA * h[n] + dtx * sBC[n];
        y += h[n] * sBC[D_STATE + n];
      }
      float zv = xz[rowt * (2 * D_INNER) + D_INNER + d];
      float sig = 1.f / (1.f + expf(-zv));
      y = (y + Dd * xv) * (zv * sig);
      xy[rowt * D_INNER + d] = y;
    }
    __syncthreads();
  }
}

// ---------------- max over L (one wave per (b, dm)) ----------------
__global__ void max_kernel(const float* __restrict__ v, float* __restrict__ ve) {
  int wave = threadIdx.x >> 5, lane = threadIdx.x & 31;
  int wg = blockIdx.x * (blockDim.x >> 5) + wave;
  if (wg >= BATCH * D_MODEL) return;
  int b = wg / D_MODEL, dm = wg % D_MODEL;
  float m = -INFINITY;
  for (int l = lane; l < SEQ; l += 32)
    m = fmaxf(m, v[(size_t)(b * SEQ + l) * D_MODEL + dm]);
#pragma unroll
  for (int o = 16; o; o >>= 1) m = fmaxf(m, __shfl_xor(m, o, 32));
  if (lane == 0) ve[wg] = m;
}

// ---------------- host launch ----------------
extern "C" void kernel_launch(void* const* d_in, const int* in_sizes, int n_in,
                              void* d_out, int out_size, void* d_ws, size_t ws_size,
                              hipStream_t stream) {
  const int*   fasta   = (const int*)d_in[0];
  const float* embedW  = (const float*)d_in[1];
  const float* in_w    = (const float*)d_in[2];
  const float* conv_w  = (const float*)d_in[3];
  const float* conv_b  = (const float*)d_in[4];
  const float* x_w     = (const float*)d_in[5];
  const float* dt_w    = (const float*)d_in[6];
  const float* dt_b    = (const float*)d_in[7];
  const float* A_log   = (const float*)d_in[8];
  const float* Dp      = (const float*)d_in[9];
  const float* out_w   = (const float*)d_in[10];
  const float* norm_w  = (const float*)d_in[11];
  const float* norm_fw = (const float*)d_in[12];
  float* out = (float*)d_out;

  // workspace layout (floats)
  float* R    = (float*)d_ws;
  float* H    = R    + (size_t)BL * D_MODEL;
  float* Nrm  = H    + (size_t)BL * D_MODEL;
  float* XZ   = Nrm  + (size_t)BL * D_MODEL;
  float* XC   = XZ   + (size_t)BL * 2 * D_INNER;
  float* DBC  = XC   + (size_t)BL * D_INNER;
  float* DT   = DBC  + (size_t)BL * DBC_W;
  float* WtIn = DT   + (size_t)BL * D_INNER;
  float* WtX  = WtIn + (size_t)IN_NP * IN_KP;
  float* WtDt = WtX  + (size_t)XP_NP * XP_KP;
  float* WtOp = WtDt + (size_t)DT_NP * DT_KP;
  float* Pbuf = WtOp + (size_t)OP_NP * OP_KP;
  float* Hbuf = Pbuf + (size_t)BATCH * NCHUNK * D_INNER * D_STATE;

  embed_kernel<<<(BL * D_MODEL + 255) / 256, 256, 0, stream>>>(fasta, embedW, H);

  for (int l = 0; l < N_LAYER; l++) {
    padT2_kernel<<<(IN_NP * IN_KP + 255) / 256, 256, 0, stream>>>(
        in_w + (size_t)l * 2 * D_INNER * D_MODEL, WtIn, 2 * D_INNER, D_MODEL, IN_NP, IN_KP);
    padT2_kernel<<<(XP_NP * XP_KP + 255) / 256, 256, 0, stream>>>(
        x_w + (size_t)l * DBC_W * D_INNER, WtX, DBC_W, D_INNER, XP_NP, XP_KP);
    padT2_kernel<<<(DT_NP * DT_KP + 255) / 256, 256, 0, stream>>>(
        dt_w + (size_t)l * D_INNER * DT_RANK, WtDt, D_INNER, DT_RANK, DT_NP, DT_KP);
    padT2_kernel<<<(OP_NP * OP_KP + 255) / 256, 256, 0, stream>>>(
        out_w + (size_t)l * D_MODEL * D_INNER, WtOp, D_MODEL, D_INNER, OP_NP, OP_KP);

    resid_rmsnorm_kernel<<<BL / 8, 256, 0, stream>>>(H, R, norm_w + l * D_MODEL, Nrm,
                                                     l == 0 ? 1 : 0);
    { // in_proj: XZ[BL,664] = Nrm @ in_w^T
      int Nt = IN_NP / 16, tiles = (BL / 16) * Nt;
      wmma_gemm_kernel<<<(tiles + 3) / 4, 128, 0, stream>>>(
          Nrm, D_MODEL, WtIn, IN_NP, XZ, 2 * D_INNER, IN_KP, Nt, tiles, nullptr, 0);
    }
    conv_silu_kernel<<<(BL * D_INNER + 255) / 256, 256, 0, stream>>>(
        XZ, XC, conv_w + l * D_INNER * D_CONV, conv_b + l * D_INNER);
    { // x_proj: DBC[BL,43] = XC @ x_w^T
      int Nt = XP_NP / 16, tiles = (BL / 16) * Nt;
      wmma_gemm_kernel<<<(tiles + 3) / 4, 128, 0, stream>>>(
          XC, D_INNER, WtX, XP_NP, DBC, DBC_W, XP_KP, Nt, tiles, nullptr, 0);
    }
    { // dt: DT[BL,332] = softplus(DBC[:, :11] @ dt_w^T + dt_b), lda=43
      int Nt = DT_NP / 16, tiles = (BL / 16) * Nt;
      wmma_gemm_kernel<<<(tiles + 3) / 4, 128, 0, stream>>>(
          DBC, DBC_W, WtDt, DT_NP, DT, D_INNER, DT_KP, Nt, tiles,
          dt_b + l * D_INNER, 1);
    }
    // chunked parallel scan: 256 blocks -> 16x the WGP coverage of a serial scan
    scan_phase1_kernel<<<BATCH * NCHUNK, 352, 0, stream>>>(
        DT, XC, DBC, A_log + (size_t)l * D_INNER * D_STATE, Pbuf, Hbuf);
    scan_phase2_kernel<<<(BATCH * D_INNER * D_STATE + 255) / 256, 256, 0, stream>>>(Pbuf, Hbuf);
    scan_phase3_kernel<<<BATCH * NCHUNK, 352, 0, stream>>>(
        DT, XC, XZ, DBC, A_log + (size_t)l * D_INNER * D_STATE, Dp + l * D_INNER, Hbuf);
    { // out_proj: H[BL,166] = y(XC) @ out_w^T
      int Nt = OP_NP / 16, tiles = (BL / 16) * Nt;
      wmma_gemm_kernel<<<(tiles + 3) / 4, 128, 0, stream>>>(
          XC, D_INNER, WtOp, OP_NP, H, D_MODEL, OP_KP, Nt, tiles, nullptr, 0);
    }
  }

  // final: residual += hidden; v = rmsnorm(residual, norm_f_w) -> out[2656:]
  resid_rmsnorm_kernel<<<BL / 8, 256, 0, stream>>>(H, R, norm_fw,
                                                   out + BATCH * D_MODEL, 0);
  // ve = max over L -> out[0:2656]
  max_kernel<<<(BATCH * D_MODEL + 7) / 8, 256, 0, stream>>>(out + BATCH * D_MODEL, out);
}